// SDecD_730144440727
// MI455X (gfx1250) — compile-verified
//
#include <hip/hip_runtime.h>

typedef float v2f __attribute__((ext_vector_type(2)));
typedef float v8f __attribute__((ext_vector_type(8)));

constexpr int CCH  = 64;
constexpr int HDIM = 512;
constexpr int W2   = 256;
constexpr int H2   = 256;
constexpr int NPIX = W2 * H2;     // 65536
constexpr int BATCH = 4;

// ---------------------------------------------------------------- K0: zero small reduction buffers
__global__ void zero_buf(float* __restrict__ p, int n) {
    int i = blockIdx.x * blockDim.x + threadIdx.x;
    if (i < n) p[i] = 0.0f;
}

__device__ __forceinline__ void haar(float e00, float e01, float e10, float e11,
                                     float& k0, float& k1, float& k2) {
    k0 = (e00 - e01 - e10 + e11) * 0.5f;   // diagonal
    k1 = (e00 - e01 + e10 - e11) * 0.5f;   // vertical
    k2 = (e00 + e01 - e10 - e11) * 0.5f;   // horizontal
}

// wave32 full reduction via lane-xor shuffles
__device__ __forceinline__ float wave_sum(float v) {
#pragma unroll
    for (int off = 16; off > 0; off >>= 1) v += __shfl_xor(v, off, 32);
    return v;
}

// block(256)-level 3-way reduction + atomic accumulate
__device__ __forceinline__ void block_accum3(float q0, float q1, float q2, float* dst) {
    __shared__ float red[3][8];
    const int lane = threadIdx.x & 31;
    const int wid  = threadIdx.x >> 5;
    q0 = wave_sum(q0); q1 = wave_sum(q1); q2 = wave_sum(q2);
    if (lane == 0) { red[0][wid] = q0; red[1][wid] = q1; red[2][wid] = q2; }
    __syncthreads();
    if (threadIdx.x == 0) {
        float t0 = 0.f, t1 = 0.f, t2 = 0.f;
#pragma unroll
        for (int i = 0; i < 8; ++i) { t0 += red[0][i]; t1 += red[1][i]; t2 += red[2][i]; }
        atomicAdd(dst + 0, t0);
        atomicAdd(dst + 1, t1);
        atomicAdd(dst + 2, t2);
    }
}

// ---------------------------------------------------------------- K1: pooled + basis sum-of-squares
__global__ void pool_basis_pass(const float* __restrict__ cen,
                                float* __restrict__ pooled,
                                float* __restrict__ sumsq) {
    const int t = threadIdx.x;                    // h in [0,256)
    const int w = blockIdx.x & (W2 - 1);
    const int c = (blockIdx.x >> 8) & (CCH - 1);
    const int b = blockIdx.x >> 14;

    const size_t base = ((size_t)(b * CCH + c) * HDIM + 2 * w) * HDIM + 2 * t;
    const float2 r0 = *reinterpret_cast<const float2*>(cen + base);
    const float2 r1 = *reinterpret_cast<const float2*>(cen + base + HDIM);

    float k0, k1, k2;
    haar(r0.x, r0.y, r1.x, r1.y, k0, k1, k2);
    pooled[(size_t)(b * CCH + c) * NPIX + w * H2 + t] =
        (r0.x + r0.y + r1.x + r1.y) * 0.25f;

    block_accum3(k0 * k0, k1 * k1, k2 * k2, sumsq + (size_t)(b * CCH + c) * 3);
}

// ---------------------------------------------------------------- K2: 1x1 conv via V_WMMA_F32_16X16X4_F32
// origins[b] (64 x 65536) = conv_w (64x64) @ pooled[b] (64x65536) + conv_b
// One wave owns ALL 4 M-tiles (64 out channels) for a 128-column group.
// A fragments (whole conv_w, 64 x v2f = 128 VGPRs) staged LDS -> registers ONCE;
// __launch_bounds__(256,1) grants the full VGPR budget so nothing spills.
// Per 16-col tile: 32-load B clause, then 64 WMMAs over 4 independent acc chains.
__global__ __launch_bounds__(256, 1)
void conv1x1_wmma(const float* __restrict__ convw,
                  const float* __restrict__ convb,
                  const float* __restrict__ pooled,
                  float* __restrict__ origins) {
    __shared__ float lds_w[CCH * CCH];   // 16 KB
    __shared__ float lds_b[CCH];
    for (int i = threadIdx.x; i < CCH * CCH; i += blockDim.x) lds_w[i] = convw[i];
    if (threadIdx.x < CCH) lds_b[threadIdx.x] = convb[threadIdx.x];
    __syncthreads();

    const int lane   = threadIdx.x & 31;
    const int gw     = blockIdx.x * (blockDim.x >> 5) + (threadIdx.x >> 5); // 0..2047
    const int ngroup = gw & 511;            // 512 groups of 128 columns
    const int b      = gw >> 9;             // batch
    const int half   = lane >> 4;           // 0: lanes 0-15, 1: lanes 16-31
    const int l16    = lane & 15;

    // Preload ALL A fragments once (16x4 f32 tile s covers K=4s..4s+3):
    // VGPR v, lane half h -> K = v + 2h, M = lane%16.
    v2f afrag[4][16];
#pragma unroll
    for (int m = 0; m < 4; ++m) {
        const int row = m * 16 + l16;
#pragma unroll
        for (int s = 0; s < 16; ++s)
            afrag[m][s] = *reinterpret_cast<const v2f*>(&lds_w[row * CCH + 4 * s + 2 * half]);
    }

    const size_t bbase = (size_t)b * CCH * NPIX;

#pragma unroll 1
    for (int nt = 0; nt < 8; ++nt) {
        const int col = ngroup * 128 + nt * 16 + l16;

        // Preload whole K=64 of B: 16 fragments, 32 independent b32 loads.
        // B fragment 4x16 f32: VGPR v, half h -> K = v + 2h, N = lane%16.
        v2f bb[16];
#pragma unroll
        for (int s = 0; s < 16; ++s) {
            const int kr = 4 * s + 2 * half;
            bb[s].x = pooled[bbase + (size_t)kr * NPIX + col];
            bb[s].y = pooled[bbase + (size_t)(kr + 1) * NPIX + col];
        }

        v8f acc[4] = {};
#pragma unroll
        for (int s = 0; s < 16; ++s) {
#pragma unroll
            for (int m = 0; m < 4; ++m) {
                acc[m] = __builtin_amdgcn_wmma_f32_16x16x4_f32(
                    false, afrag[m][s], false, bb[s], (short)0, acc[m], false, false);
            }
        }

#pragma unroll
        for (int m = 0; m < 4; ++m) {
#pragma unroll
            for (int i = 0; i < 8; ++i) {
                const int o = m * 16 + i + 8 * half;     // D row M = i + 8*half
                origins[bbase + (size_t)o * NPIX + col] = acc[m][i] + lds_b[o];
            }
        }
    }
}

// ---------------------------------------------------------------- K3: dot[b,c,k] = sum_n origins * basis
__global__ void dot_pass(const float* __restrict__ cen,
                         const float* __restrict__ origins,
                         float* __restrict__ dotb) {
    const int t = threadIdx.x;
    const int w = blockIdx.x & (W2 - 1);
    const int c = (blockIdx.x >> 8) & (CCH - 1);
    const int b = blockIdx.x >> 14;

    const size_t base = ((size_t)(b * CCH + c) * HDIM + 2 * w) * HDIM + 2 * t;
    const float2 r0 = *reinterpret_cast<const float2*>(cen + base);
    const float2 r1 = *reinterpret_cast<const float2*>(cen + base + HDIM);
    float k0, k1, k2;
    haar(r0.x, r0.y, r1.x, r1.y, k0, k1, k2);

    const float og = origins[(size_t)(b * CCH + c) * NPIX + w * H2 + t];

    block_accum3(og * k0, og * k1, og * k2, dotb + (size_t)(b * CCH + c) * 3);
}

// ---------------------------------------------------------------- K4: out_n = sum_k coef_k * basis_k(n)
// coef_k = dot_k / max(sqrt(sumsq_k), 1e-12)^2  (both normalizations folded)
__global__ void out_pass(const float* __restrict__ cen,
                         const float* __restrict__ sumsq,
                         const float* __restrict__ dotb,
                         float* __restrict__ out) {
    const int t = threadIdx.x;
    const int w = blockIdx.x & (W2 - 1);
    const int c = (blockIdx.x >> 8) & (CCH - 1);
    const int b = blockIdx.x >> 14;

    const size_t bc = (size_t)(b * CCH + c);
    float coef[3];
#pragma unroll
    for (int k = 0; k < 3; ++k) {
        const float m = fmaxf(sqrtf(sumsq[bc * 3 + k]), 1e-12f);
        coef[k] = dotb[bc * 3 + k] / (m * m);
    }

    const size_t base = (bc * HDIM + 2 * w) * HDIM + 2 * t;
    const float2 r0 = *reinterpret_cast<const float2*>(cen + base);
    const float2 r1 = *reinterpret_cast<const float2*>(cen + base + HDIM);
    float k0, k1, k2;
    haar(r0.x, r0.y, r1.x, r1.y, k0, k1, k2);

    out[bc * NPIX + w * H2 + t] = coef[0] * k0 + coef[1] * k1 + coef[2] * k2;
}

// ----------------------------------------------------------------
extern "C" void kernel_launch(void* const* d_in, const int* in_sizes, int n_in,
                              void* d_out, int out_size, void* d_ws, size_t ws_size,
                              hipStream_t stream) {
    const float* cen    = (const float*)d_in[0];  // (4,64,512,512)
    const float* conv_w = (const float*)d_in[1];  // (64,64)
    const float* conv_b = (const float*)d_in[2];  // (64,)
    float* out = (float*)d_out;                   // (4,64,256,256)

    float* ws      = (float*)d_ws;
    float* pooled  = ws;                                      // 16,777,216 floats
    float* origins = pooled + (size_t)BATCH * CCH * NPIX;     // 16,777,216 floats
    float* sumsq   = origins + (size_t)BATCH * CCH * NPIX;    // 768 floats
    float* dotb    = sumsq + BATCH * CCH * 3;                 // 768 floats

    // K0: zero the reduction buffers (sumsq ++ dot are contiguous: 1536 floats)
    zero_buf<<<6, 256, 0, stream>>>(sumsq, BATCH * CCH * 3 * 2);

    // K1: pooled + sum of squares of basis
    pool_basis_pass<<<BATCH * CCH * W2, 256, 0, stream>>>(cen, pooled, sumsq);

    // K2: 1x1 conv via f32 WMMA. 2048 waves = 4 b * 512 ngroups; 8 waves/block
    conv1x1_wmma<<<256, 256, 0, stream>>>(conv_w, conv_b, pooled, origins);

    // K3: dot(origins, basis)
    dot_pass<<<BATCH * CCH * W2, 256, 0, stream>>>(cen, origins, dotb);

    // K4: final combine
    out_pass<<<BATCH * CCH * W2, 256, 0, stream>>>(cen, sumsq, dotb, out);
}